// BoidsODE_28338194219042
// MI455X (gfx1250) — compile-verified
//
#include <hip/hip_runtime.h>

// ---------------------------------------------------------------------------
// BoidsODE edge-message + segment_sum for MI455X (gfx1250).
// Gather/scatter streaming kernel: edge_index (51.2MB, read-once) is
// double-buffered through LDS with CDNA5 async global->LDS copies using
// non-temporal hints; node arrays (2.4MB) and the output (800KB) stay
// L2/WGP$-resident; scatter uses f32 global atomics.
// ---------------------------------------------------------------------------

#define BLOCK 256
#define EPT   4                   // edges per thread per stage (one B128 each)
#define TILE  (BLOCK * EPT)       // 1024 edges per stage

__device__ __constant__ float cA1 = 5e-06f;
__device__ __constant__ float cA2 = 0.0005f;
__device__ __constant__ float cA3 = 1e-08f;

#if __has_builtin(__builtin_amdgcn_s_wait_asynccnt)
#define WAIT_ASYNCCNT(n) __builtin_amdgcn_s_wait_asynccnt(n)
#else
#define WAIT_ASYNCCNT(n) asm volatile("s_wait_asynccnt %0" :: "n"(n))
#endif

// Async copy of 16 bytes/lane from global memory into LDS.
// vdst operand = 32-bit LDS byte address (low 32 bits of the generic pointer
// to a __shared__ object are the LDS offset on amdgcn); vaddr = 64-bit global
// address. Tracked by ASYNCcnt, not LOADcnt. TH_LOAD_NT: the edge stream is
// read exactly once -> keep it from displacing the L2-resident node arrays.
__device__ __forceinline__ void async_copy_b128_nt(const void* gsrc, void* lds_dst) {
  unsigned ldsOff = (unsigned)(unsigned long long)lds_dst;
  asm volatile("global_load_async_to_lds_b128 %0, %1, off th:TH_LOAD_NT"
               :
               : "v"(ldsOff), "v"((unsigned long long)gsrc)
               : "memory");
}

__device__ __forceinline__ void process_edge(
    int i, int j,
    const float2* __restrict__ pos, const float2* __restrict__ vel,
    const float* __restrict__ field, const int* __restrict__ ptype,
    const float4* ptab, float* __restrict__ out)
{
  float2 pi = pos[i];
  float2 pj = pos[j];
  float2 vi = vel[i];
  float2 vj = vel[j];
  float dpx = pj.x - pi.x;
  float dpy = pj.y - pi.y;
  float dvx = vj.x - vi.x;
  float dvy = vj.y - vi.y;
  float d2  = dpx * dpx + dpy * dpy;
  float safe = (d2 > 0.0f) ? d2 : 1.0f;
  float4 p = ptab[ptype[i]];                 // one ds_read_b128
  // cohesion + separation share the dp direction:
  //   p0*A1*dp - p2*A3*dp/safe  =  dp * (p0*A1 - p2*A3/safe)
  float coef = p.x * cA1 - (p.z * cA3) / safe;
  float aln  = p.y * cA2;
  float f    = field[j];
  float mx = (coef * dpx + aln * dvx) * f;
  float my = (coef * dpy + aln * dvy) * f;
  atomicAdd(&out[2 * i + 0], mx);
  atomicAdd(&out[2 * i + 1], my);
}

__global__ void __launch_bounds__(BLOCK) boids_zero_kernel(float* __restrict__ out, int n) {
  int i = blockIdx.x * blockDim.x + threadIdx.x;
  if (i < n) out[i] = 0.0f;
}

__global__ void __launch_bounds__(BLOCK) boids_edge_kernel(
    const float2* __restrict__ pos, const float2* __restrict__ vel,
    const float* __restrict__ p_table, const float* __restrict__ field,
    const int* __restrict__ ptype,
    const int* __restrict__ e_dst, const int* __restrict__ e_src,
    float* __restrict__ out, int E)
{
  __shared__ __align__(16) int dstBuf[2][TILE];
  __shared__ __align__(16) int srcBuf[2][TILE];
  __shared__ float4 ptab[16];

  const int tid = threadIdx.x;
  if (tid < 16) {
    ptab[tid] = make_float4(p_table[3 * tid + 0],
                            p_table[3 * tid + 1],
                            p_table[3 * tid + 2], 0.0f);
  }
  __syncthreads();

  const int tiles   = (E + TILE - 1) / TILE;
  const int tstride = gridDim.x;
  int tile = blockIdx.x;

  // Prologue: kick off async copies for this block's first (full) tile.
  if (tile < tiles) {
    const int base = tile * TILE;
    if (base + TILE <= E) {
      async_copy_b128_nt(e_dst + base + tid * EPT, &dstBuf[0][tid * EPT]);
      async_copy_b128_nt(e_src + base + tid * EPT, &srcBuf[0][tid * EPT]);
    }
  }

  int buf = 0;
  for (; tile < tiles; tile += tstride, buf ^= 1) {
    const int  base     = tile * TILE;
    const bool fullCur  = (base + TILE <= E);
    const int  ntile    = tile + tstride;
    const int  nbase    = ntile * TILE;
    const bool fullNext = (ntile < tiles) && (nbase + TILE <= E);

    // Issue next tile's async copies (uniform branch -> EXEC all ones),
    // then wait until only those 2 remain outstanding: current tile ready.
    if (fullNext) {
      async_copy_b128_nt(e_dst + nbase + tid * EPT, &dstBuf[buf ^ 1][tid * EPT]);
      async_copy_b128_nt(e_src + nbase + tid * EPT, &srcBuf[buf ^ 1][tid * EPT]);
      WAIT_ASYNCCNT(2);
    } else {
      WAIT_ASYNCCNT(0);
    }
    asm volatile("" ::: "memory");   // LDS staging now visible to this wave

    if (fullCur) {
      // Each lane consumes exactly the 16B it async-loaded: no barrier needed.
      int4 d4 = *(const int4*)&dstBuf[buf][tid * EPT];
      int4 s4 = *(const int4*)&srcBuf[buf][tid * EPT];
      process_edge(d4.x, s4.x, pos, vel, field, ptype, ptab, out);
      process_edge(d4.y, s4.y, pos, vel, field, ptype, ptab, out);
      process_edge(d4.z, s4.z, pos, vel, field, ptype, ptab, out);
      process_edge(d4.w, s4.w, pos, vel, field, ptype, ptab, out);
    } else {
      // Tail tile (not hit for E = 6.4M, kept for safety): direct global path.
#pragma unroll
      for (int k = 0; k < EPT; ++k) {
        int e = base + tid * EPT + k;
        if (e < E) {
          int di = __builtin_nontemporal_load(e_dst + e);
          int sj = __builtin_nontemporal_load(e_src + e);
          process_edge(di, sj, pos, vel, field, ptype, ptab, out);
        }
      }
    }
  }
}

extern "C" void kernel_launch(void* const* d_in, const int* in_sizes, int n_in,
                              void* d_out, int out_size, void* d_ws, size_t ws_size,
                              hipStream_t stream) {
  const float2* pos     = (const float2*)d_in[0];   // [N,2] f32
  const float2* vel     = (const float2*)d_in[1];   // [N,2] f32
  const float*  p_table = (const float*)d_in[2];    // [16,3] f32
  const float*  field   = (const float*)d_in[3];    // [N,1] f32
  const int*    ptype   = (const int*)d_in[4];      // [N] i32
  const int*    eidx    = (const int*)d_in[5];      // [2,E] i32: row0=dst, row1=src

  const int E = in_sizes[5] / 2;
  const int* e_dst = eidx;
  const int* e_src = eidx + E;
  float* out = (float*)d_out;                       // [N,2] f32

  // Zero the (poisoned) output, then scatter-accumulate edge messages.
  boids_zero_kernel<<<(out_size + BLOCK - 1) / BLOCK, BLOCK, 0, stream>>>(out, out_size);

  const int tiles = (E + TILE - 1) / TILE;          // 6250 for E = 6.4M
  int grid = tiles < 1024 ? tiles : 1024;           // persistent blocks, ~6 tiles each
  boids_edge_kernel<<<grid, BLOCK, 0, stream>>>(pos, vel, p_table, field, ptype,
                                                e_dst, e_src, out, E);
}